// SALayer_83717502534258
// MI455X (gfx1250) — compile-verified
//
#include <hip/hip_runtime.h>
#include <hip/hip_bf16.h>
#include <math.h>

typedef __attribute__((ext_vector_type(2))) float v2f;
typedef __attribute__((ext_vector_type(4))) float v4f;
typedef __attribute__((ext_vector_type(8))) float v8f;

#define C_FEAT 128
#define K_OFF  27

// ---------------------------------------------------------------------------
// Kernel 0: zero the conv accumulator (N+1 floats; harness poisons ws)
// ---------------------------------------------------------------------------
__global__ __launch_bounds__(256) void sal_zero_kernel(float* __restrict__ conv, int n) {
    int i = blockIdx.x * blockDim.x + threadIdx.x;
    if (i < n) conv[i] = 0.0f;
}

// ---------------------------------------------------------------------------
// Kernel 1: channel pool.  One wave32 per 16-voxel tile.
//   mean: V_WMMA_F32_16X16X4_F32 with B = ones (exact f32 row-sum on the
//         matrix pipe), 32 accumulating WMMAs over K=128.
//   max : VALU on the same loaded values; lane l covers half the columns of
//         row (l&15); shfl_xor(16) merges the halves.
// A-matrix 16x4 f32 layout (ISA 7.12.2): lane l -> row M=l&15,
//   VGPR0 = K { 0 | 2 } for halves {0|1}, VGPR1 = K { 1 | 3 }.
// ---------------------------------------------------------------------------
__global__ __launch_bounds__(256) void sal_pool_kernel(const float* __restrict__ feat,
                                                       float2* __restrict__ f2,
                                                       int n_vox, int n_tiles) {
    const int lane = threadIdx.x & 31;
    const int wave = threadIdx.x >> 5;
    const int tile = blockIdx.x * 8 + wave;       // 8 waves/block, 16 voxels/wave
    if (tile >= n_tiles) return;                  // wave-uniform -> EXEC stays full

    const int li   = lane & 15;                   // row within tile
    const int half = lane >> 4;                   // which K pair we own
    const int base = tile * 16;
    const int lrow = min(base + li, n_vox - 1);   // clamp (dup rows; stores predicated)
    const float* rp = feat + (size_t)lrow * C_FEAT + 2 * half;

    v8f acc = {};                                 // D: every column = row-sum
    float vmax = -INFINITY;
    const v2f ones = {1.0f, 1.0f};
#pragma unroll 4
    for (int kk = 0; kk < C_FEAT; kk += 4) {
        v2f a;
        a.x = rp[kk];                             // col kk + 2*half
        a.y = rp[kk + 1];                         // col kk + 2*half + 1
        vmax = fmaxf(vmax, fmaxf(a.x, a.y));
        acc = __builtin_amdgcn_wmma_f32_16x16x4_f32(
            /*neg_a=*/false, a, /*neg_b=*/false, ones,
            /*c_mod=*/(short)0, acc, /*reuse_a=*/false, /*reuse_b=*/false);
    }
    // merge column halves of the max (all lanes participate, uniform)
    vmax = fmaxf(vmax, __shfl_xor(vmax, 16, 32));

    // D layout: VGPR m holds row (m + 8*half) at N=li -> lane with li<8 owns
    // rows {8*half .. 8*half+7} in acc[0..7]; select element li.
    float s = acc[0];
#pragma unroll
    for (int m = 1; m < 8; ++m)
        if (li == m) s = acc[m];

    if (li < 8) {
        int r = base + half * 8 + li;
        if (r < n_vox) f2[r].x = s * (1.0f / (float)C_FEAT);   // mean
    }
    if (half == 0) {
        int r = base + li;
        if (r < n_vox) f2[r].y = vmax;                          // max
    }
}

// ---------------------------------------------------------------------------
// Kernel 2: pair gather -> 2-term dot -> scatter-add.  4 pairs per thread via
// int4 (b128) index loads; weight table (54 f32) staged in LDS; f2/conv are
// L2-resident (4MB / 2MB).  Padding pairs hit dummy slot conv[N].
// ---------------------------------------------------------------------------
__global__ __launch_bounds__(256) void sal_pairs_kernel(const int4* __restrict__ k4,
                                                        const int4* __restrict__ i4,
                                                        const int4* __restrict__ o4,
                                                        const float* __restrict__ wg,
                                                        const float2* __restrict__ f2,
                                                        float* __restrict__ conv,
                                                        int n_quads) {
    __shared__ float w[K_OFF * 2];
    if (threadIdx.x < K_OFF * 2) w[threadIdx.x] = wg[threadIdx.x];
    __syncthreads();

    int q = blockIdx.x * blockDim.x + threadIdx.x;
    int stride = gridDim.x * blockDim.x;
    for (; q < n_quads; q += stride) {
        int4 k = k4[q];
        int4 i = i4[q];
        int4 o = o4[q];
        float2 fa = f2[i.x], fb = f2[i.y], fc = f2[i.z], fd = f2[i.w];
        float ca = fmaf(fa.x, w[2 * k.x], fa.y * w[2 * k.x + 1]);
        float cb = fmaf(fb.x, w[2 * k.y], fb.y * w[2 * k.y + 1]);
        float cc = fmaf(fc.x, w[2 * k.z], fc.y * w[2 * k.z + 1]);
        float cd = fmaf(fd.x, w[2 * k.w], fd.y * w[2 * k.w + 1]);
        atomicAdd(&conv[o.x], ca);
        atomicAdd(&conv[o.y], cb);
        atomicAdd(&conv[o.z], cc);
        atomicAdd(&conv[o.w], cd);
    }
}

// ---------------------------------------------------------------------------
// Kernel 3: out = features * sigmoid(conv[row]).  One float4 per thread,
// nontemporal (TH=NT) on the 512MB feature/out streams so conv/f2 stay in L2.
// conv[row] is wave-uniform per 32-lane group -> hardware broadcast load.
// ---------------------------------------------------------------------------
__global__ __launch_bounds__(256) void sal_scale_kernel(const float* __restrict__ feat,
                                                        const float* __restrict__ conv,
                                                        float* __restrict__ out,
                                                        int n_vec) {
    int gid = blockIdx.x * blockDim.x + threadIdx.x;   // float4 index
    if (gid >= n_vec) return;
    int row = gid >> 5;                                // 32 float4 per 128-ch row
    float x = conv[row];
    float g = 1.0f / (1.0f + __expf(-x));
    v4f v = __builtin_nontemporal_load((const v4f*)feat + gid);
    v4f r = {v.x * g, v.y * g, v.z * g, v.w * g};
    __builtin_nontemporal_store(r, (v4f*)out + gid);
}

// ---------------------------------------------------------------------------
extern "C" void kernel_launch(void* const* d_in, const int* in_sizes, int n_in,
                              void* d_out, int out_size, void* d_ws, size_t ws_size,
                              hipStream_t stream) {
    const float* features = (const float*)d_in[0];   // (N, 128) f32
    const float* weight   = (const float*)d_in[1];   // (27, 2, 1) f32 -> 54 floats
    const int*   kidx     = (const int*)d_in[2];     // (P,)
    const int*   in_idx   = (const int*)d_in[3];     // (P,)
    const int*   out_idx  = (const int*)d_in[4];     // (P,)
    float*       out      = (float*)d_out;

    const int n_vox = in_sizes[0] / C_FEAT;          // 500000
    const int p_max = in_sizes[2];                   // 27 * N = 13.5M (mult of 4)

    // workspace: f2 (N float2) | conv (N+1 float)
    float2* f2   = (float2*)d_ws;
    float*  conv = (float*)d_ws + 2 * (size_t)n_vox;

    // 0) zero accumulator (incl. dummy padding slot)
    {
        int n = n_vox + 1;
        sal_zero_kernel<<<(n + 255) / 256, 256, 0, stream>>>(conv, n);
    }
    // 1) channel pool (WMMA mean + VALU max)
    {
        int n_tiles = (n_vox + 15) / 16;
        int blocks  = (n_tiles + 7) / 8;             // 8 tiles (waves) per block
        sal_pool_kernel<<<blocks, 256, 0, stream>>>(features, f2, n_vox, n_tiles);
    }
    // 2) gather/contract/scatter-add
    {
        int n_quads = p_max / 4;
        int blocks  = (n_quads + 255) / 256;
        if (blocks > 16384) blocks = 16384;          // grid-stride
        sal_pairs_kernel<<<blocks, 256, 0, stream>>>((const int4*)kidx,
                                                     (const int4*)in_idx,
                                                     (const int4*)out_idx,
                                                     weight, f2, conv, n_quads);
    }
    // 3) sigmoid gate * features
    {
        int n_vec  = n_vox * (C_FEAT / 4);
        int blocks = (n_vec + 255) / 256;
        sal_scale_kernel<<<blocks, 256, 0, stream>>>(features, conv, out, n_vec);
    }
    (void)n_in; (void)out_size; (void)ws_size;
}